// TreeSoftmax_1803886264584
// MI455X (gfx1250) — compile-verified
//
#include <hip/hip_runtime.h>
#include <stdint.h>

// Tree softmax over fixed 4-ary tree, N=145 nodes -> 144 channels.
// x: [B=4, C=144, H=256, W=256] f32, out same shape.
// Sibling group g (g=0..35): channels 4g..4g+3, parent node g.
// res[node] = p[node] * res[parent], BFS order; parents are nodes 1..35.

#define TS_C    144
#define TS_HW   65536                 // 256*256
#define TS_BLK  128                   // pixels (= threads) per block, 4 waves
#define TS_NBPB (TS_HW / TS_BLK)      // 512 blocks per batch image
#define TS_GRP  36
#define TS_LOG2E 1.4426950408889634f

__global__ __launch_bounds__(TS_BLK) void tree_softmax_async_kernel(
    const float* __restrict__ x, float* __restrict__ out)
{
    __shared__ __align__(16) float smem[TS_C * TS_BLK];   // 72 KB

    const int tid  = threadIdx.x;
    const int lane = tid & 31;
    const int wv   = tid >> 5;                            // wave id: 0..3

    const int b       = (int)blockIdx.x / TS_NBPB;
    const int hwStart = ((int)blockIdx.x % TS_NBPB) * TS_BLK;

    const uint64_t planeBase = (uint64_t)b * (uint64_t)TS_C * (uint64_t)TS_HW
                             + (uint64_t)hwStart;         // element offset of (b, c=0, hwStart)
    const uint64_t gbase = (uint64_t)(uintptr_t)(x + planeBase);
    // Low 32 bits of generic pointer to LDS == LDS byte offset of smem.
    const unsigned sbase  = (unsigned)(uintptr_t)(&smem[0]);
    const unsigned lane16 = (unsigned)lane * 16u;         // b128: 16 bytes per lane

    // ---- Async DMA stage: wave wv copies channels {wv, wv+4, wv+8, ...}.
    // One global_load_async_to_lds_b128 per channel slice:
    //   32 lanes x 16B = 512B = 128 floats = this block's slice of one channel plane.
    // Streamed once, no reuse -> non-temporal.
#pragma unroll
    for (int i = 0; i < TS_C / 4; ++i) {
        const int c = 4 * i + wv;
        const unsigned lofs = sbase + (unsigned)c * (TS_BLK * 4u) + lane16;
        const unsigned gofs = (unsigned)c * (TS_HW * 4u) + lane16;
        asm volatile("global_load_async_to_lds_b128 %0, %1, %2 th:TH_LOAD_NT"
                     :: "v"(lofs), "v"(gofs), "s"(gbase)
                     : "memory");
    }
    asm volatile("s_wait_asynccnt 0" ::: "memory");
    __syncthreads();

    // ---- Tree softmax: 36 sibling groups of 4 contiguous channels.
    float res[35];                                        // final prob of node (idx+1)
    float* __restrict__ outp = out + planeBase + (uint64_t)tid;

#pragma unroll
    for (int g = 0; g < TS_GRP; ++g) {
        const float v0 = smem[(4 * g + 0) * TS_BLK + tid];
        const float v1 = smem[(4 * g + 1) * TS_BLK + tid];
        const float v2 = smem[(4 * g + 2) * TS_BLK + tid];
        const float v3 = smem[(4 * g + 3) * TS_BLK + tid];

        // exp(v - m) == exp2(v*log2e - m*log2e): one FMA feeding v_exp_f32
        const float m  = fmaxf(fmaxf(v0, v1), fmaxf(v2, v3));
        const float mh = m * TS_LOG2E;
        const float e0 = __builtin_amdgcn_exp2f(__builtin_fmaf(v0, TS_LOG2E, -mh));
        const float e1 = __builtin_amdgcn_exp2f(__builtin_fmaf(v1, TS_LOG2E, -mh));
        const float e2 = __builtin_amdgcn_exp2f(__builtin_fmaf(v2, TS_LOG2E, -mh));
        const float e3 = __builtin_amdgcn_exp2f(__builtin_fmaf(v3, TS_LOG2E, -mh));

        // parent node g: final prob already computed (g < 4g+1), root => 1.0
        const float pr = (g == 0) ? 1.0f : res[g - 1];
        const float sc = pr * __builtin_amdgcn_rcpf(e0 + e1 + e2 + e3);

        const float r0 = e0 * sc;
        const float r1 = e1 * sc;
        const float r2 = e2 * sc;
        const float r3 = e3 * sc;

        // keep results only for nodes that are parents later (nodes <= 35)
        if (4 * g + 1 <= 35) res[4 * g + 0] = r0;
        if (4 * g + 2 <= 35) res[4 * g + 1] = r1;
        if (4 * g + 3 <= 35) res[4 * g + 2] = r2;
        if (4 * g + 4 <= 35) res[4 * g + 3] = r3;

        // streamed once -> non-temporal stores (th:TH_STORE_NT)
        __builtin_nontemporal_store(r0, outp + (size_t)(4 * g + 0) * TS_HW);
        __builtin_nontemporal_store(r1, outp + (size_t)(4 * g + 1) * TS_HW);
        __builtin_nontemporal_store(r2, outp + (size_t)(4 * g + 2) * TS_HW);
        __builtin_nontemporal_store(r3, outp + (size_t)(4 * g + 3) * TS_HW);
    }
}

extern "C" void kernel_launch(void* const* d_in, const int* in_sizes, int n_in,
                              void* d_out, int out_size, void* d_ws, size_t ws_size,
                              hipStream_t stream)
{
    const float* x  = (const float*)d_in[0];
    float* out      = (float*)d_out;

    const long long total = (long long)in_sizes[0];              // B*144*65536
    const int B = (int)(total / ((long long)TS_C * (long long)TS_HW));

    dim3 grid((unsigned)(B * TS_NBPB));
    dim3 block(TS_BLK);
    tree_softmax_async_kernel<<<grid, block, 0, stream>>>(x, out);

    (void)n_in; (void)out_size; (void)d_ws; (void)ws_size;
}